// pool_22874995818908
// MI455X (gfx1250) — compile-verified
//
#include <hip/hip_runtime.h>

typedef __attribute__((ext_vector_type(2))) float v2f;
typedef __attribute__((ext_vector_type(8))) float v8f;

#define BATCH 8
#define SEQ   4096
#define DIM   1024
#define CR    64            // rows per chunk
#define NC    (SEQ / CR)    // 64 chunks
#define D4    (DIM / 4)     // 256 float4 columns

// ---------------------------------------------------------------------------
// Kernel 1: per-(b, chunk) column sums over 64 rows, coalesced float4 stream.
// ws layout: float ws[b][chunk][DIM]  (2 MB total)
// Regular-temporal loads on purpose: this pass warms the 192 MB L2 with all
// 128 MB of x so the scan pass re-reads from L2 instead of HBM.
// ---------------------------------------------------------------------------
__global__ void cummean_chunksum(const float* __restrict__ x,
                                 float4* __restrict__ ws) {
  int f  = blockIdx.x * blockDim.x + threadIdx.x;  // BATCH*NC*D4 threads
  int d4 = f & (D4 - 1);
  int bc = f >> 8;                                 // b*NC + chunk
  const float4* xp = (const float4*)x + (size_t)bc * CR * D4 + d4;
  float4 s = make_float4(0.f, 0.f, 0.f, 0.f);
#pragma unroll 4
  for (int r = 0; r < CR; ++r) {
    if (r + 8 < CR) __builtin_prefetch(xp + (size_t)(r + 8) * D4, 0, 0);
    float4 v = xp[(size_t)r * D4];
    s.x += v.x; s.y += v.y; s.z += v.z; s.w += v.w;
  }
  ws[(size_t)bc * D4 + d4] = s;
}

// Fast ~1-ulp reciprocal: v_rcp_f32 + one Newton-Raphson step.
__device__ __forceinline__ float fast_rcp(float x) {
  float inv = __builtin_amdgcn_rcpf(x);
  float e   = __builtin_fmaf(-x, inv, 1.0f);
  return __builtin_fmaf(e, inv, inv);
}

// ---------------------------------------------------------------------------
// Kernel 2: one wave per (b, chunk, 16-col d-tile).
// 16-row inclusive prefix-sum tile = L(16x16 lower-triangular ones) * X(16x16),
// computed as 4 chained V_WMMA_F32_16X16X4_F32 (K-slices of 4).
// Carry between 16-row sub-tiles = D row 15 (VGPR7, lanes 16-31), broadcast.
// Output stores are non-temporal so 'out' (128 MB) doesn't evict x from L2.
// ---------------------------------------------------------------------------
__global__ void cummean_wmma_scan(const float* __restrict__ x,
                                  const float* __restrict__ ps,
                                  float* __restrict__ out) {
  int gid  = blockIdx.x * blockDim.x + threadIdx.x;
  int wave = gid >> 5;
  int lane = threadIdx.x & 31;
  int dt = wave & (DIM / 16 - 1);   // 64 d-tiles
  int c  = (wave >> 6) & (NC - 1);  // chunk  (wave-uniform)
  int b  = wave >> 12;              // batch
  int h  = lane >> 4;               // lane half
  int n  = lane & 15;               // column within tile / matrix row M
  int dcol = dt * 16 + n;

  // Exclusive chunk offset for this column: sum partials of preceding chunks
  // (2 MB array, L2-resident).
  float off = 0.f;
  const float* pp = ps + (size_t)b * NC * DIM + dcol;
  for (int cc = 0; cc < c; ++cc) off += pp[(size_t)cc * DIM];

  // A-matrix K-slices: A_k[m, j+2h] = (m >= 4k + j + 2h) ? 1 : 0
  // (16x4 f32 A layout: lane = M, VGPR j + half h -> K = j + 2h)
  v2f a[4];
#pragma unroll
  for (int k = 0; k < 4; ++k) {
#pragma unroll
    for (int j = 0; j < 2; ++j) {
      int kk = 4 * k + j + 2 * h;
      a[k][j] = (n >= kk) ? 1.0f : 0.0f;
    }
  }

  v8f carry;
#pragma unroll
  for (int r = 0; r < 8; ++r) carry[r] = off;

  const size_t base = ((size_t)b * SEQ + (size_t)c * CR) * DIM + dcol;
  const float* xb = x + base;
  float*       ob = out + base;

#pragma unroll
  for (int t = 0; t < 4; ++t) {
    int s0 = c * CR + t * 16;
    const float* xt = xb + (size_t)t * 16 * DIM;

    // B-matrix 4x16 slices: VGPR j + half h -> K row = j + 2h
    v2f bk[4];
#pragma unroll
    for (int k = 0; k < 4; ++k)
#pragma unroll
      for (int j = 0; j < 2; ++j)
        bk[k][j] = xt[(size_t)(4 * k + j + 2 * h) * DIM];

    v8f acc = carry;
    acc = __builtin_amdgcn_wmma_f32_16x16x4_f32(false, a[0], false, bk[0],
                                                (short)0, acc, false, false);
    acc = __builtin_amdgcn_wmma_f32_16x16x4_f32(false, a[1], false, bk[1],
                                                (short)0, acc, false, false);
    acc = __builtin_amdgcn_wmma_f32_16x16x4_f32(false, a[2], false, bk[2],
                                                (short)0, acc, false, false);
    acc = __builtin_amdgcn_wmma_f32_16x16x4_f32(false, a[3], false, bk[3],
                                                (short)0, acc, false, false);

    // Scale by 1/(s+1) and store NT: VGPR r, half h -> global row s0 + r + 8h
    float* ot = ob + (size_t)t * 16 * DIM;
#pragma unroll
    for (int r = 0; r < 8; ++r) {
      int   row = s0 + r + 8 * h;
      float inv = fast_rcp((float)(row + 1));
      __builtin_nontemporal_store(acc[r] * inv, ot + (size_t)(r + 8 * h) * DIM);
    }

    // Carry = D row 15 (already includes previous carry): VGPR7, lane 16+n.
    float bs = __shfl(acc[7], 16 + n, 32);
#pragma unroll
    for (int r = 0; r < 8; ++r) carry[r] = bs;
  }
}

// ---------------------------------------------------------------------------
extern "C" void kernel_launch(void* const* d_in, const int* in_sizes, int n_in,
                              void* d_out, int out_size, void* d_ws,
                              size_t ws_size, hipStream_t stream) {
  (void)in_sizes; (void)n_in; (void)out_size; (void)ws_size;
  const float* x  = (const float*)d_in[0];
  float*      out = (float*)d_out;
  float*      ws  = (float*)d_ws;  // needs BATCH*NC*DIM*4 = 2 MB

  int th1 = BATCH * NC * D4;                 // 131072 threads
  cummean_chunksum<<<th1 / 256, 256, 0, stream>>>(x, (float4*)ws);

  int waves = BATCH * NC * (DIM / 16);       // 32768 waves
  cummean_wmma_scan<<<waves * 32 / 256, 256, 0, stream>>>(x, ws, out);
}